// Model_60713657696924
// MI455X (gfx1250) — compile-verified
//
#include <hip/hip_runtime.h>
#include <hip/hip_bf16.h>

typedef __attribute__((ext_vector_type(2))) float v2f;
typedef __attribute__((ext_vector_type(4))) float v4f;
typedef __attribute__((ext_vector_type(8))) float v8f;
typedef unsigned int u32x4 __attribute__((ext_vector_type(4)));
typedef int          i32x4 __attribute__((ext_vector_type(4)));
typedef int          i32x8 __attribute__((ext_vector_type(8)));

#define BM 128
#define BN 128
#define BK 32
#define LDA 36     // As row stride (floats): BK + 4 pad (TDM: pad_interval=32dw, pad_amount=4dw)
#define KPAIRS 16  // BK/2 interleaved k-pair rows for B
#define LDBP 288   // Bs pair-row stride (floats): 256 + 32 pad; %4==0 (b128), %64==32 (bank split)

#define NTHREADS 256
#define KDIM 2048
#define NDIM 2048
#define NGROUPS 8

#if __has_builtin(__builtin_amdgcn_tensor_load_to_lds)
#define HAVE_TDM 1
#else
#define HAVE_TDM 0
#endif

struct GroupMeta {
    int m;                      // rows in this group
    int tileBase;               // first global tile id of this group
    int mTiles;
    int pad_;
    unsigned long long aOff, bOff, cOff;  // element offsets into flat buffers
};

struct Params {
    GroupMeta g[NGROUPS];
    int totalTiles;
};

__global__ __launch_bounds__(NTHREADS)
void ragged_gemm_wmma_f32(const float* __restrict__ A,
                          const float* __restrict__ B,
                          const float* __restrict__ C,
                          const float* __restrict__ alpha,
                          const float* __restrict__ beta,
                          float* __restrict__ Out,
                          Params p)
{
    __shared__ float As[2][BM * LDA];       // 2 * 18432 B
    __shared__ float Bs[2][KPAIRS * LDBP];  // 2 * 18432 B

    const int bid = blockIdx.x;

    // --- locate ragged group (8 groups, unrolled scan) ---
    int g = 0;
#pragma unroll
    for (int i = 1; i < NGROUPS; ++i)
        if (bid >= p.g[i].tileBase) g = i;
    const GroupMeta gm = p.g[g];

    const int nTilesN = NDIM / BN;                 // 16
    const int local  = bid - gm.tileBase;
    const int mBase  = (local / nTilesN) * BM;
    const int nBase  = (local % nTilesN) * BN;
    const int M      = gm.m;

    const float* __restrict__ Ag = A + gm.aOff;
    const float* __restrict__ Bg = B + gm.bOff;
    const float* __restrict__ Cg = C + gm.cOff;
    float* __restrict__       Og = Out + gm.cOff;

    const int tid     = threadIdx.x;
    const int lane    = tid & 31;
    const int wid     = tid >> 5;
    const int waveM   = wid & 3;        // 4 wave-rows  * 32 rows = 128
    const int waveN   = wid >> 2;       // 2 wave-cols  * 64 cols = 128
    const int l16     = lane & 15;
    const int halfSel = lane >> 4;      // 0: lanes 0-15, 1: lanes 16-31

    v8f acc[2][4];
#pragma unroll
    for (int i = 0; i < 2; ++i)
#pragma unroll
        for (int j = 0; j < 4; ++j)
            acc[i][j] = (v8f){0.f, 0.f, 0.f, 0.f, 0.f, 0.f, 0.f, 0.f};

#if HAVE_TDM
    // --- A stager via Tensor Data Mover: one DMA per stage, issued by wave 0.
    //     2D descriptor: tile BK x BM from tensor (K-stride rows), LDS padding
    //     reproduces LDA=36 (pad 4 dwords every 32 dwords), OOB rows read as 0
    //     (handles the ragged M edge natively). ---
    const unsigned rowsLeft = (unsigned)(M - mBase);
    auto issueTdmA = [&](int buf, int k0) {
        const unsigned long long ga =
            (unsigned long long)(const void*)(Ag + (size_t)mBase * KDIM + k0);
        const unsigned lds = (unsigned)(size_t)(const void*)&As[buf][0]; // flat->LDS low 32 bits

        u32x4 g0;
        g0.x = 1u;                                            // count=1, user descriptor
        g0.y = lds;                                           // lds_addr (bytes)
        g0.z = (unsigned)(ga & 0xFFFFFFFFu);                  // global_addr[31:0]
        g0.w = (unsigned)((ga >> 32) & 0x01FFFFFFu)           // global_addr[56:32]
             | (2u << 30);                                    // type = 2 ("image")

        i32x8 g1;
        g1[0] = (int)((2u << 16)        // data_size = 4 bytes
                    | (1u << 20)        // pad_enable
                    | (4u << 22)        // pad_interval: every 32 dwords
                    | (3u << 25));      // pad_amount: 4 dwords
        g1[1] = (int)(((unsigned)KDIM & 0xFFFFu) << 16);          // tensor_dim0[15:0]
        g1[2] = (int)(((unsigned)KDIM >> 16)                      // tensor_dim0[31:16]
                    | ((rowsLeft & 0xFFFFu) << 16));              // tensor_dim1[15:0]
        g1[3] = (int)((rowsLeft >> 16)                            // tensor_dim1[31:16]
                    | ((unsigned)BK << 16));                      // tile_dim0 = 32
        g1[4] = (int)BM;                                          // tile_dim1 = 128, tile_dim2 = 0
        g1[5] = (int)KDIM;                                        // tensor_dim0_stride[31:0]
        g1[6] = 0;                                                // stride hi / dim1_stride (unused, 2D)
        g1[7] = 0;

        const i32x4 z4 = (i32x4){0, 0, 0, 0};
        const i32x8 z8 = (i32x8){0, 0, 0, 0, 0, 0, 0, 0};
        // amdgpu-toolchain (clang-23) 6-arg form: (g0, g1, g2, g3, extra_group, cpol)
        __builtin_amdgcn_tensor_load_to_lds(g0, g1, z4, z4, z8, 0);
    };
#else
    // --- fallback A stager: float4 global loads -> b128 LDS stores ---
    auto loadA = [&](int buf, int k0) {
        const int c4 = (tid & 7) * 4;
        const int r0 = tid >> 3;
#pragma unroll
        for (int rp = 0; rp < 4; ++rp) {
            const int row  = r0 + rp * 32;
            const int grow = mBase + row;
            v4f v = (v4f){0.f, 0.f, 0.f, 0.f};
            if (grow < M)
                v = *(const v4f*)(Ag + (size_t)grow * KDIM + (k0 + c4));
            *(v4f*)(&As[buf][row * LDA + c4]) = v;
        }
    };
#endif

    // --- B stager: k-pair interleaved layout Bs[kp][2*n + q] = B[k0+2*kp+q][nBase+n]
    //     so a WMMA B fragment is ONE aligned b64 LDS load. ---
    auto loadB = [&](int buf, int k0) {
        const int n4 = (tid & 31) * 4;      // n offset, step 4 (0..124)
        const int kp0 = tid >> 5;           // 0..7
#pragma unroll
        for (int pass = 0; pass < 2; ++pass) {
            const int kp = kp0 + pass * 8;  // 0..15
            const float* r0 = Bg + (size_t)(k0 + 2 * kp)     * NDIM + (nBase + n4);
            const float* r1 = Bg + (size_t)(k0 + 2 * kp + 1) * NDIM + (nBase + n4);
            const v4f v0 = *(const v4f*)r0;
            const v4f v1 = *(const v4f*)r1;
            v4f lo = (v4f){v0.x, v1.x, v0.y, v1.y};
            v4f hi = (v4f){v0.z, v1.z, v0.w, v1.w};
            float* dst = &Bs[buf][kp * LDBP + 2 * n4];
            *(v4f*)(dst)     = lo;
            *(v4f*)(dst + 4) = hi;
        }
    };

#if HAVE_TDM
    if (wid == 0) issueTdmA(0, 0);
#else
    loadA(0, 0);
#endif
    loadB(0, 0);
#if HAVE_TDM
    __builtin_amdgcn_s_wait_tensorcnt(0);   // no-op for waves with TENSORcnt==0
#endif
    __syncthreads();

    int buf = 0;
    for (int k0 = 0; k0 < KDIM; k0 += BK) {
        const int nk = k0 + BK;
        if (nk < KDIM) {        // stage k0+BK into the back buffer
#if HAVE_TDM
            if (wid == 0) issueTdmA(buf ^ 1, nk);
#else
            loadA(buf ^ 1, nk);
#endif
            loadB(buf ^ 1, nk);
        }

        // L2 warm-up two stages ahead for B (global_prefetch_b8)
        const int pk = k0 + 2 * BK;
        if (pk < KDIM)
            __builtin_prefetch(Bg + (size_t)(pk + (tid >> 5)) * NDIM + nBase + (tid & 31) * 4, 0, 3);

#pragma unroll
        for (int kk = 0; kk < BK; kk += 4) {
            const int ko = kk + 2 * halfSel;     // K pair selected by lane half
            const int kp = ko >> 1;              // pair-row in interleaved Bs

            // A fragment 16x4: lane m = M%16, b64 load of A[m][ko..ko+1]
            v2f afrag[2];
#pragma unroll
            for (int i = 0; i < 2; ++i) {
                const int m = waveM * 32 + i * 16 + l16;
                afrag[i] = *(const v2f*)(&As[buf][m * LDA + ko]);
            }

            // B fragment 4x16: lane n = N%16, single b64 load of interleaved pair
            v2f bfrag[4];
#pragma unroll
            for (int j = 0; j < 4; ++j) {
                const int n = waveN * 64 + j * 16 + l16;
                bfrag[j] = *(const v2f*)(&Bs[buf][kp * LDBP + 2 * n]);
            }

#pragma unroll
            for (int i = 0; i < 2; ++i)
#pragma unroll
                for (int j = 0; j < 4; ++j)
                    acc[i][j] = __builtin_amdgcn_wmma_f32_16x16x4_f32(
                        /*neg_a=*/false, afrag[i],
                        /*neg_b=*/false, bfrag[j],
                        /*c_mod=*/(short)0, acc[i][j],
                        /*reuse_a=*/false, /*reuse_b=*/false);
        }

#if HAVE_TDM
        __builtin_amdgcn_s_wait_tensorcnt(0);   // publish DMA'd A tile before barrier
#endif
        __syncthreads();
        buf ^= 1;
    }

    // --- epilogue: out = alpha * (A@B) + beta * C ---
    const float al = alpha[g];
    const float be = beta[g];
#pragma unroll
    for (int i = 0; i < 2; ++i) {
        const int rowBase = mBase + waveM * 32 + i * 16 + 8 * halfSel;
#pragma unroll
        for (int j = 0; j < 4; ++j) {
            const int col = nBase + waveN * 64 + j * 16 + l16;
#pragma unroll
            for (int v = 0; v < 8; ++v) {
                const int row = rowBase + v;     // VGPR v: M = v (+8 for lane half 1)
                if (row < M) {
                    const size_t idx = (size_t)row * NDIM + col;
                    Og[idx] = al * acc[i][j][v] + be * Cg[idx];
                }
            }
        }
    }
}

extern "C" void kernel_launch(void* const* d_in, const int* in_sizes, int n_in,
                              void* d_out, int out_size, void* d_ws, size_t ws_size,
                              hipStream_t stream)
{
    // Static ragged metadata (matches reference M_LIST/K_LIST/N_LIST)
    static const int M_LIST[NGROUPS] = {4096, 2048, 1024, 3072, 2048, 1024, 1536, 1584};

    Params p;
    unsigned long long aOff = 0, bOff = 0, cOff = 0;
    int tileBase = 0;
    for (int i = 0; i < NGROUPS; ++i) {
        const int m  = M_LIST[i];
        const int mT = (m + BM - 1) / BM;
        p.g[i].m        = m;
        p.g[i].mTiles   = mT;
        p.g[i].pad_     = 0;
        p.g[i].tileBase = tileBase;
        p.g[i].aOff     = aOff;
        p.g[i].bOff     = bOff;
        p.g[i].cOff     = cOff;
        tileBase += mT * (NDIM / BN);
        aOff += (unsigned long long)m * KDIM;
        bOff += (unsigned long long)KDIM * NDIM;
        cOff += (unsigned long long)m * NDIM;
    }
    p.totalTiles = tileBase;   // 129 * 16 = 2064 workgroups

    const float* A     = (const float*)d_in[0];
    const float* B     = (const float*)d_in[1];
    const float* C     = (const float*)d_in[2];
    const float* alpha = (const float*)d_in[3];
    const float* beta  = (const float*)d_in[4];
    float* Out         = (float*)d_out;

    dim3 grid(p.totalTiles), block(NTHREADS);
    hipLaunchKernelGGL(ragged_gemm_wmma_f32, grid, block, 0, stream,
                       A, B, C, alpha, beta, Out, p);
}